// EGNN_40321152974877
// MI455X (gfx1250) — compile-verified
//
#include <hip/hip_runtime.h>

// ---------------------------------------------------------------------------
// EGNN (GCNII-style) forward for MI455X / gfx1250.
//   - dense GEMMs via V_WMMA_F32_16X16X4_F32, LDS-staged A tiles, pair-packed B
//   - sparse aggregation via L2-resident atomic scatter (working set < 192MB L2)
// ---------------------------------------------------------------------------

#define N_NODES   50000
#define N_EDGES   1600000
#define HDIM      128
#define N_CLASSES 40
#define NC_PAD    48            // 40 padded to 3 column tiles
#define N_LAYERS  8

// combination coefficients: BETA=0.1, RESIDUAL_WEIGHT=0.1 -> agg coef 0.8
#define C_AGG  0.8f
#define C_RES  0.1f
#define C_X0   0.1f

typedef __attribute__((ext_vector_type(2))) float v2f;
typedef __attribute__((ext_vector_type(8))) float v8f;

// ---------------------------------------------------------------------------
// Pair-packed B layout: Bp[(p*ldn + n)*2 + {0,1}] = B[2p][n], B[2p+1][n]
// A lane's B fragment for K-step k0 is VGPR0->K=k0+2*half, VGPR1->K=k0+2*half+1
// -> pair index p = (k0>>1)+half, one contiguous 8-byte load.
// ---------------------------------------------------------------------------

// WMMA 16x16 f32 tile, A fragments from LDS (row-major 16x128 tile),
// B fragments from pair-packed global. K fixed at 128. No guards anywhere.
__device__ __forceinline__ v8f egnn_wmma_tile_lds(const float* __restrict__ As,
                                                  const float* __restrict__ Bp,
                                                  int ldn, int colBase) {
  const int lane = threadIdx.x & 31;
  const int half = lane >> 4;
  const int mn   = lane & 15;
  const float* aPtr = As + mn * HDIM + half * 2;            // + k0 per step
  const float* bPtr = Bp + ((size_t)half * ldn + colBase + mn) * 2;
  v8f acc = {0.f, 0.f, 0.f, 0.f, 0.f, 0.f, 0.f, 0.f};
#pragma unroll 8
  for (int k0 = 0; k0 < HDIM; k0 += 4) {
    v2f a = *(const v2f*)(aPtr + k0);
    v2f b = *(const v2f*)(bPtr + (size_t)(k0 >> 1) * ldn * 2);
    // 8 args: (neg_a, A, neg_b, B, c_mod, C, reuse_a, reuse_b)
    acc = __builtin_amdgcn_wmma_f32_16x16x4_f32(false, a, false, b,
                                                (short)0, acc, false, false);
  }
  return acc;
}

// Cooperative copy of one contiguous 16x128 A tile (8KB) into LDS.
// Valid because lda == HDIM: the tile is a single contiguous block.
__device__ __forceinline__ void egnn_stage_a(const float* __restrict__ A,
                                             int rowBase, float* __restrict__ As) {
  const float4* src = (const float4*)(A + (size_t)rowBase * HDIM);
  float4* dst = (float4*)As;
  dst[threadIdx.x]       = src[threadIdx.x];
  dst[threadIdx.x + 256] = src[threadIdx.x + 256];
  __syncthreads();
}

// input layer: h = relu(X @ W_in + b_in); also copies into x0.
// 256 threads = 8 waves, wave w computes column tile w.
__global__ void egnn_gemm_in(const float* __restrict__ X, const float* __restrict__ Bp,
                             const float* __restrict__ bias,
                             float* __restrict__ hA, float* __restrict__ x0) {
  __shared__ float As[16 * HDIM];
  const int rowBase = blockIdx.x * 16;
  egnn_stage_a(X, rowBase, As);
  const int colBase = (threadIdx.x >> 5) * 16;
  v8f acc = egnn_wmma_tile_lds(As, Bp, HDIM, colBase);
  const int lane = threadIdx.x & 31;
  const int half = lane >> 4;
  const int n = colBase + (lane & 15);
  const float bn = bias[n];
#pragma unroll
  for (int v = 0; v < 8; ++v) {
    const int m = rowBase + v + 8 * half;       // C/D layout: VGPR v -> M = v + 8*half
    float z = fmaxf(acc[v] + bn, 0.f);
    hA[(size_t)m * HDIM + n] = z;
    x0[(size_t)m * HDIM + n] = z;
  }
}

// hidden layer: h = srelu(hB @ W, sb) = relu(z - sb) + sb
__global__ void egnn_gemm_srelu(const float* __restrict__ hB, const float* __restrict__ Bp,
                                const float* __restrict__ sb, float* __restrict__ hA) {
  __shared__ float As[16 * HDIM];
  const int rowBase = blockIdx.x * 16;
  egnn_stage_a(hB, rowBase, As);
  const int colBase = (threadIdx.x >> 5) * 16;
  v8f acc = egnn_wmma_tile_lds(As, Bp, HDIM, colBase);
  const int lane = threadIdx.x & 31;
  const int half = lane >> 4;
  const int n = colBase + (lane & 15);
  const float b = sb[n];
#pragma unroll
  for (int v = 0; v < 8; ++v) {
    const int m = rowBase + v + 8 * half;
    hA[(size_t)m * HDIM + n] = fmaxf(acc[v] - b, 0.f) + b;
  }
}

// output layer: out = h @ W_out + b_out. One wave per block, grid (3125, 3).
// B comes zero-padded to 48 cols, so only the store is guarded.
__global__ void egnn_gemm_out(const float* __restrict__ h, const float* __restrict__ Bp,
                              const float* __restrict__ bias, float* __restrict__ out) {
  const int rowBase = blockIdx.x * 16;
  const int colBase = blockIdx.y * 16;
  const int lane = threadIdx.x & 31;
  const int half = lane >> 4;
  const int mn   = lane & 15;
  const float* aPtr = h + (size_t)(rowBase + mn) * HDIM + half * 2;
  const float* bPtr = Bp + ((size_t)half * NC_PAD + colBase + mn) * 2;
  v8f acc = {0.f, 0.f, 0.f, 0.f, 0.f, 0.f, 0.f, 0.f};
#pragma unroll 8
  for (int k0 = 0; k0 < HDIM; k0 += 4) {
    v2f a = *(const v2f*)(aPtr + k0);
    v2f b = *(const v2f*)(bPtr + (size_t)(k0 >> 1) * NC_PAD * 2);
    acc = __builtin_amdgcn_wmma_f32_16x16x4_f32(false, a, false, b,
                                                (short)0, acc, false, false);
  }
  const int n = colBase + mn;
  if (n < N_CLASSES) {
    const float bn = bias[n];
#pragma unroll
    for (int v = 0; v < 8; ++v) {
      const int m = rowBase + v + 8 * half;
      out[(size_t)m * N_CLASSES + n] = acc[v] + bn;
    }
  }
}

// ---------------------------------------------------------------------------
// Weight packing (once per launch)
// ---------------------------------------------------------------------------
// L consecutive 128x128 matrices -> pair-packed [L][64][128][2]
__global__ void egnn_pack_w128(const float* __restrict__ W, float* __restrict__ Bp,
                               int total /* L*64*128 */) {
  int idx = blockIdx.x * blockDim.x + threadIdx.x;
  if (idx >= total) return;
  int l = idx >> 13;                 // / (64*128)
  int r = idx & 8191;
  int kk = r >> 7;
  int n  = r & 127;
  const float* src = W + (size_t)l * HDIM * HDIM;
  float2 p;
  p.x = src[(size_t)(2 * kk)     * HDIM + n];
  p.y = src[(size_t)(2 * kk + 1) * HDIM + n];
  *(float2*)(Bp + (size_t)idx * 2) = p;
}

// 128x40 -> pair-packed zero-padded [64][48][2]
__global__ void egnn_pack_wout(const float* __restrict__ W, float* __restrict__ Bp) {
  int idx = blockIdx.x * blockDim.x + threadIdx.x;
  if (idx >= 64 * NC_PAD) return;
  int kk = idx / NC_PAD;
  int n  = idx % NC_PAD;
  float2 p;
  p.x = (n < N_CLASSES) ? W[(size_t)(2 * kk)     * N_CLASSES + n] : 0.f;
  p.y = (n < N_CLASSES) ? W[(size_t)(2 * kk + 1) * N_CLASSES + n] : 0.f;
  *(float2*)(Bp + (size_t)idx * 2) = p;
}

// ---------------------------------------------------------------------------
// GCN normalization
// ---------------------------------------------------------------------------
__global__ void egnn_deg_init(float* __restrict__ deg) {
  int i = blockIdx.x * blockDim.x + threadIdx.x;
  if (i < N_NODES) deg[i] = 1.0f;                      // fresh self-loop weight
}

__global__ void egnn_deg_accum(const long long* __restrict__ ei, float* __restrict__ deg) {
  int e = blockIdx.x * blockDim.x + threadIdx.x;
  if (e >= N_EDGES) return;
  long long r = ei[e];
  long long c = ei[(size_t)N_EDGES + e];
  if (r != c) atomicAdd(&deg[r], 1.0f);                // existing self-loops weight 0
}

__global__ void egnn_dinv(float* __restrict__ deg) {    // in-place deg -> deg^-1/2
  int i = blockIdx.x * blockDim.x + threadIdx.x;
  if (i < N_NODES) deg[i] = rsqrtf(deg[i]);            // deg >= 1 always
}

__global__ void egnn_edge_norm(const long long* __restrict__ ei,
                               const float* __restrict__ dinv, float* __restrict__ norm) {
  int e = blockIdx.x * blockDim.x + threadIdx.x;
  if (e >= N_EDGES) return;
  long long r = ei[e];
  long long c = ei[(size_t)N_EDGES + e];
  norm[e] = (r == c) ? 0.f : dinv[r] * dinv[c];
}

// ---------------------------------------------------------------------------
// Per-layer sparse aggregation
// ---------------------------------------------------------------------------
// agg[i] = dinv[i]^2 * h[i]   (the appended self-loop contribution)
__global__ void egnn_agg_init(const float* __restrict__ h, const float* __restrict__ dinv,
                              float* __restrict__ agg) {
  int gid = blockIdx.x * blockDim.x + threadIdx.x;     // one float4 quad per thread
  if (gid >= N_NODES * (HDIM / 4)) return;
  int i = gid >> 5;
  int q = gid & 31;
  float s = dinv[i];
  s = s * s;
  const float4 v = *(const float4*)(h + (size_t)i * HDIM + q * 4);
  float4 r;
  r.x = s * v.x; r.y = s * v.y; r.z = s * v.z; r.w = s * v.w;
  *(float4*)(agg + (size_t)i * HDIM + q * 4) = r;
}

// one wave per edge: agg[col] += norm[e] * h[row]; float4 per lane
__global__ void egnn_scatter(const long long* __restrict__ ei, const float* __restrict__ norm,
                             const float* __restrict__ h, float* __restrict__ agg) {
  int e = blockIdx.x * (blockDim.x >> 5) + (threadIdx.x >> 5);
  if (e >= N_EDGES) return;
  float w = norm[e];                                   // uniform across the wave
  if (w == 0.f) return;                                // dropped self-loops
  int lane = threadIdx.x & 31;
  long long r = ei[e];
  long long c = ei[(size_t)N_EDGES + e];
  const float4 hv = *(const float4*)(h + (size_t)r * HDIM + lane * 4);
  float* dst = agg + (size_t)c * HDIM + lane * 4;
  atomicAdd(dst + 0, w * hv.x);                        // result unused -> no-return atomic
  atomicAdd(dst + 1, w * hv.y);
  atomicAdd(dst + 2, w * hv.z);
  atomicAdd(dst + 3, w * hv.w);
}

// hB = 0.8*agg + 0.1*hA + 0.1*x0
__global__ void egnn_combine(const float* __restrict__ agg, const float* __restrict__ hA,
                             const float* __restrict__ x0, float* __restrict__ hB) {
  int gid = blockIdx.x * blockDim.x + threadIdx.x;
  if (gid >= N_NODES * (HDIM / 4)) return;
  size_t off = (size_t)gid * 4;
  const float4 a = *(const float4*)(agg + off);
  const float4 h = *(const float4*)(hA + off);
  const float4 x = *(const float4*)(x0 + off);
  float4 r;
  r.x = C_AGG * a.x + C_RES * h.x + C_X0 * x.x;
  r.y = C_AGG * a.y + C_RES * h.y + C_X0 * x.y;
  r.z = C_AGG * a.z + C_RES * h.z + C_X0 * x.z;
  r.w = C_AGG * a.w + C_RES * h.w + C_X0 * x.w;
  *(float4*)(hB + off) = r;
}

// ---------------------------------------------------------------------------
extern "C" void kernel_launch(void* const* d_in, const int* in_sizes, int n_in,
                              void* d_out, int out_size, void* d_ws, size_t ws_size,
                              hipStream_t stream) {
  const float*     x       = (const float*)d_in[0];
  const long long* ei      = (const long long*)d_in[1];   // int64 [2, N_EDGES]
  const float*     W_in    = (const float*)d_in[2];
  const float*     b_in    = (const float*)d_in[3];
  const float*     gcn_W   = (const float*)d_in[4];       // [8,128,128]
  const float*     srelu_b = (const float*)d_in[5];       // [8,128]
  const float*     W_out   = (const float*)d_in[6];       // [128,40]
  const float*     b_out   = (const float*)d_in[7];
  float*           out     = (float*)d_out;

  // workspace layout (floats), all 16B-aligned
  float* ws   = (float*)d_ws;
  float* dinv = ws;                                       //    50,048 (padded)
  float* norm = dinv + 50048;                             // 1,600,000
  float* x0   = norm + N_EDGES;                           // 6,400,000
  float* hA   = x0 + (size_t)N_NODES * HDIM;              // 6,400,000
  float* hB   = hA + (size_t)N_NODES * HDIM;              // 6,400,000
  float* agg  = hB + (size_t)N_NODES * HDIM;              // 6,400,000
  float* WpI  = agg + (size_t)N_NODES * HDIM;             //    16,384
  float* WpL  = WpI + HDIM * HDIM;                        //   131,072
  float* WpO  = WpL + (size_t)N_LAYERS * HDIM * HDIM;     //     6,144

  const int nodeBlocks = (N_NODES + 255) / 256;
  const int edgeBlocks = (N_EDGES + 255) / 256;
  const int quadBlocks = (N_NODES * (HDIM / 4) + 255) / 256;

  // 0) pack weights into pair-packed B layout (one b64 load per WMMA fragment)
  egnn_pack_w128<<<(64 * 128 + 255) / 256, 256, 0, stream>>>(W_in, WpI, 64 * 128);
  egnn_pack_w128<<<(N_LAYERS * 64 * 128 + 255) / 256, 256, 0, stream>>>(
      gcn_W, WpL, N_LAYERS * 64 * 128);
  egnn_pack_wout<<<(64 * NC_PAD + 255) / 256, 256, 0, stream>>>(W_out, WpO);

  // 1) normalization
  egnn_deg_init <<<nodeBlocks, 256, 0, stream>>>(dinv);
  egnn_deg_accum<<<edgeBlocks, 256, 0, stream>>>(ei, dinv);
  egnn_dinv     <<<nodeBlocks, 256, 0, stream>>>(dinv);
  egnn_edge_norm<<<edgeBlocks, 256, 0, stream>>>(ei, dinv, norm);

  // 2) input layer (50000 = 3125*16 exactly -> no row guards); 8 waves/block,
  //    one column tile per wave, A tile staged in LDS
  egnn_gemm_in<<<N_NODES / 16, 256, 0, stream>>>(x, WpI, b_in, hA, x0);

  // 3) 8 propagation layers
  for (int l = 0; l < N_LAYERS; ++l) {
    egnn_agg_init<<<quadBlocks, 256, 0, stream>>>(hA, dinv, agg);
    egnn_scatter <<<(N_EDGES + 7) / 8, 256, 0, stream>>>(ei, norm, hA, agg);
    egnn_combine <<<quadBlocks, 256, 0, stream>>>(agg, hA, x0, hB);
    egnn_gemm_srelu<<<N_NODES / 16, 256, 0, stream>>>(
        hB, WpL + (size_t)l * HDIM * HDIM, srelu_b + (size_t)l * HDIM, hA);
  }

  // 4) output layer, N=40 padded to 3 column tiles, store-guarded
  dim3 gOut(N_NODES / 16, NC_PAD / 16);
  egnn_gemm_out<<<gOut, 32, 0, stream>>>(hA, WpO, b_out, out);
}